// NODE_REN_9010841387093
// MI455X (gfx1250) — compile-verified
//
#include <hip/hip_runtime.h>
#include <math.h>

// ---------------------------------------------------------------------------
// NODE-REN forward on MI455X (gfx1250, wave32).
//
// All fp32 (matches reference numerics through the tanh recurrence).
// Matrix work on V_WMMA_F32_16X16X4_F32; LDS tiles stored k-pair-major so
// every WMMA fragment is a single aligned ds_load_b64 into the source pair.
//
//   1. P  = 0.5*Pstar*Pstar^T + eps*I        (WMMA)
//   2. H  = X*X^T + eps*I                    (WMMA)
//   3. derive Y, D11, C1, RHSB (elementwise)
//   4. Cholesky: P = L*L^T                   (single-WG right-looking)
//   5. A  = P^-1 * Y                         (column-parallel trsm x2)
//   6. B1 = P^-1 * (-H2-Chi)
//   7. Vx = xi * C1^T                        (WMMA)
//   8. scan: 16 block-steps of 64 cols; cross-block prefix via WMMA GEMM,
//      in-block 64x64 strictly-lower triangle scalar in registers
//   9. out = xi*A^T + w*B1^T                 (WMMA x2)
// ---------------------------------------------------------------------------

typedef __attribute__((ext_vector_type(2))) float v2f;
typedef __attribute__((ext_vector_type(8))) float v8f;

#define NXC 1024
#define NQC 1024
#define BC  8192
#define NHC 2048
#define EPSC 0.001f
#define KB 64            // scan block width

#define TM 128           // block tile M
#define TN 128           // block tile N
#define TK 32            // block tile K
#define TKP (TK / 2)     // k-pairs per tile
#define PADP 16          // v2f padding: shifts lane16-31 half onto other banks

// C[M x N] = alpha * A[M x K] * B[N x K]^T  (+ Cin) (+ diag_eps on diagonal)
// 8 waves: 4 along M (32 rows each) x 2 along N (64 cols each);
// each wave: 2x4 subtiles of 16x16 -> 8 WMMAs per 4-deep k-step.
__global__ __launch_bounds__(256) void gemm_nt_f32_wmma(
    const float* __restrict__ A, int lda,
    const float* __restrict__ B, int ldb,
    const float* __restrict__ Cin, int ldcin,
    float* __restrict__ C, int ldc,
    int M, int N, int K, float alpha, float diag_eps)
{
    // k-pair-major tiles: sAp[p][m] = {A[m][2p], A[m][2p+1]} within the tile.
    __shared__ v2f sAp[TKP][TM + PADP];
    __shared__ v2f sBp[TKP][TN + PADP];

    const int tid  = threadIdx.x;
    const int lane = tid & 31;
    const int wave = tid >> 5;
    const int wm   = wave & 3;    // 4 waves along M, 32 rows each
    const int wn   = wave >> 2;   // 2 waves along N, 64 cols each
    const int lr   = lane & 15;
    const int hi   = lane >> 4;   // lane half: selects K pair 0 / 1 of the step
    const int gm0  = blockIdx.y * TM;
    const int gn0  = blockIdx.x * TN;

    v8f acc[2][4] = {};

    for (int k0 = 0; k0 < K; k0 += TK) {
        // Stage tiles: each thread moves aligned float2's (global b64 loads).
        for (int idx = tid; idx < TM * TKP; idx += 256) {
            const int p = idx & (TKP - 1), m = idx >> 4;
            const int gm = gm0 + m, gk = k0 + 2 * p;
            v2f av = {0.0f, 0.0f};
            if (gm < M && gk < K) av = *(const v2f*)(A + (size_t)gm * lda + gk);
            sAp[p][m] = av;
        }
        for (int idx = tid; idx < TN * TKP; idx += 256) {
            const int p = idx & (TKP - 1), n = idx >> 4;
            const int gn = gn0 + n, gk = k0 + 2 * p;
            v2f bv = {0.0f, 0.0f};
            if (gn < N && gk < K) bv = *(const v2f*)(B + (size_t)gn * ldb + gk);
            sBp[p][n] = bv;
        }
        __syncthreads();

#pragma unroll
        for (int kp = 0; kp < TKP; kp += 2) {   // each iter = 4 K-depth
            v2f af[2], bf[4];
#pragma unroll
            for (int s = 0; s < 2; ++s)
                af[s] = sAp[kp + hi][wm * 32 + s * 16 + lr];
#pragma unroll
            for (int s = 0; s < 4; ++s)
                bf[s] = sBp[kp + hi][wn * 64 + s * 16 + lr];
#pragma unroll
            for (int si = 0; si < 2; ++si)
#pragma unroll
                for (int sj = 0; sj < 4; ++sj)
                    acc[si][sj] = __builtin_amdgcn_wmma_f32_16x16x4_f32(
                        false, af[si], false, bf[sj], (short)0, acc[si][sj],
                        false, false);
        }
        __syncthreads();
    }

    // Epilogue. C/D layout: VGPR r, lanes 0-15 -> M=r, lanes 16-31 -> M=r+8.
#pragma unroll
    for (int si = 0; si < 2; ++si) {
#pragma unroll
        for (int sj = 0; sj < 4; ++sj) {
            const int rbase = gm0 + wm * 32 + si * 16 + hi * 8;
            const int col   = gn0 + wn * 64 + sj * 16 + lr;
            if (col < N) {
#pragma unroll
                for (int r = 0; r < 8; ++r) {
                    const int row = rbase + r;
                    if (row < M) {
                        float v = alpha * acc[si][sj][r];
                        if (Cin) v += Cin[(size_t)row * ldcin + col];
                        if (diag_eps != 0.0f && row == col) v += diag_eps;
                        C[(size_t)row * ldc + col] = v;
                    }
                }
            }
        }
    }
}

// Elementwise derivation of Y, D11, C1, RHSB from H / Y1 / Chi. lam folded in.
__global__ __launch_bounds__(256) void derive_kernel(
    const float* __restrict__ H, const float* __restrict__ Y1,
    const float* __restrict__ Chi,
    float* __restrict__ Yd, float* __restrict__ D11,
    float* __restrict__ C1, float* __restrict__ RHSB)
{
    const int j = blockIdx.x * blockDim.x + threadIdx.x;
    const int i = blockIdx.y;
    const float lam_i = 0.5f * H[(size_t)(NXC + i) * NHC + NXC + i];
    const float h1 = H[(size_t)i * NHC + j];
    const float h2 = H[(size_t)i * NHC + NXC + j];
    const float h4 = H[(size_t)(NXC + i) * NHC + NXC + j];
    Yd[(size_t)i * NXC + j]   = -0.5f * (h1 + Y1[(size_t)i * NXC + j] - Y1[(size_t)j * NXC + i]);
    D11[(size_t)i * NQC + j]  = (j < i) ? (-h4 / lam_i) : 0.0f;   // -tril(H4,-1)/lam
    C1[(size_t)i * NXC + j]   = Chi[(size_t)j * NQC + i] / lam_i; // Chi^T / lam
    RHSB[(size_t)i * NQC + j] = -h2 - Chi[(size_t)i * NQC + j];
}

// In-place right-looking Cholesky of SPD P (lower factor in lower triangle).
__global__ __launch_bounds__(1024) void cholesky_kernel(float* __restrict__ P, int N)
{
    __shared__ float sdiag;
    for (int k = 0; k < N; ++k) {
        if (threadIdx.x == 0) sdiag = sqrtf(P[(size_t)k * N + k]);
        __syncthreads();
        const float d = sdiag;
        for (int i = k + 1 + threadIdx.x; i < N; i += blockDim.x)
            P[(size_t)i * N + k] /= d;
        if (threadIdx.x == 0) P[(size_t)k * N + k] = d;
        __syncthreads();
        const int rem = N - k - 1;
        for (int idx = threadIdx.x; idx < rem * rem; idx += blockDim.x) {
            const int ii = idx / rem, jj = idx % rem;
            if (jj <= ii) {
                const int i = k + 1 + ii, j = k + 1 + jj;
                P[(size_t)i * N + j] -= P[(size_t)i * N + k] * P[(size_t)j * N + k];
            }
        }
        __syncthreads();
    }
}

// Solve L * Z = Y, one thread per RHS column (coalesced across columns).
__global__ __launch_bounds__(256) void trsm_lower(
    const float* __restrict__ L, const float* __restrict__ Y,
    float* __restrict__ Z, int N, int C)
{
    const int c = blockIdx.x * blockDim.x + threadIdx.x;
    if (c >= C) return;
    for (int i = 0; i < N; ++i) {
        float s = Y[(size_t)i * C + c];
        for (int j = 0; j < i; ++j) s -= L[(size_t)i * N + j] * Z[(size_t)j * C + c];
        Z[(size_t)i * C + c] = s / L[(size_t)i * N + i];
    }
}

// Solve L^T * Xo = Z.
__global__ __launch_bounds__(256) void trsm_upper(
    const float* __restrict__ L, const float* __restrict__ Z,
    float* __restrict__ Xo, int N, int C)
{
    const int c = blockIdx.x * blockDim.x + threadIdx.x;
    if (c >= C) return;
    for (int i = N - 1; i >= 0; --i) {
        float s = Z[(size_t)i * C + c];
        for (int j = i + 1; j < N; ++j) s -= L[(size_t)j * N + i] * Xo[(size_t)j * C + c];
        Xo[(size_t)i * C + c] = s / L[(size_t)i * N + i];
    }
}

// In-block tanh recurrence over 64 columns. Cross-block prefix is in Gacc
// (WMMA GEMM); in-block strictly-lower 64x64 triangle done here with w in
// registers and the D11 tile broadcast from LDS.
__global__ __launch_bounds__(256) void scan_block_kernel(
    const float* __restrict__ Vx, const float* __restrict__ Gacc, int has_G,
    const float* __restrict__ D11, float* __restrict__ w, int t)
{
    __shared__ float sD[KB][KB + 1];
    const int col0 = t * KB;
    for (int idx = threadIdx.x; idx < KB * KB; idx += blockDim.x) {
        const int i = idx >> 6, j = idx & 63;
        sD[i][j] = D11[(size_t)(col0 + i) * NQC + (col0 + j)];
    }
    __syncthreads();
    const int b = blockIdx.x * blockDim.x + threadIdx.x;
    float wl[KB];
#pragma unroll
    for (int i = 0; i < KB; ++i) {
        float v = Vx[(size_t)b * NQC + col0 + i];
        if (has_G) v += Gacc[(size_t)b * KB + i];
        for (int j = 0; j < i; ++j) v += wl[j] * sD[i][j];
        wl[i] = tanhf(v);
    }
#pragma unroll
    for (int i = 0; i < KB; ++i)
        w[(size_t)b * NQC + col0 + i] = wl[i];
}

extern "C" void kernel_launch(void* const* d_in, const int* in_sizes, int n_in,
                              void* d_out, int out_size, void* d_ws, size_t ws_size,
                              hipStream_t stream)
{
    const float* xi    = (const float*)d_in[0];   // [8192 x 1024]
    const float* Pstar = (const float*)d_in[1];   // [1024 x 1024]
    const float* Chi   = (const float*)d_in[2];   // [1024 x 1024]
    const float* Y1    = (const float*)d_in[3];   // [1024 x 1024]
    const float* X     = (const float*)d_in[4];   // [2048 x 2048]
    float* out = (float*)d_out;                   // [8192 x 1024]

    // Workspace carve-up (~120 MB of fp32).
    float* W = (float*)d_ws;
    size_t o = 0;
    float* Hm   = W + o; o += (size_t)NHC * NHC;
    float* P    = W + o; o += (size_t)NXC * NXC;
    float* Yd   = W + o; o += (size_t)NXC * NXC;
    float* D11  = W + o; o += (size_t)NQC * NQC;
    float* C1   = W + o; o += (size_t)NQC * NXC;
    float* RHSB = W + o; o += (size_t)NXC * NQC;
    float* Amat = W + o; o += (size_t)NXC * NXC;
    float* B1   = W + o; o += (size_t)NXC * NQC;
    float* Zt   = W + o; o += (size_t)NXC * NXC;
    float* Vx   = W + o; o += (size_t)BC * NQC;
    float* w    = W + o; o += (size_t)BC * NQC;
    float* Gacc = W + o; o += (size_t)BC * KB;

    auto ggrid = [](int M, int N) { return dim3((N + TN - 1) / TN, (M + TM - 1) / TM); };

    // 1. P = 0.5 * Pstar Pstar^T + eps I
    gemm_nt_f32_wmma<<<ggrid(NXC, NXC), 256, 0, stream>>>(
        Pstar, NXC, Pstar, NXC, nullptr, 0, P, NXC, NXC, NXC, NXC, 0.5f, EPSC);
    // 2. H = X X^T + eps I
    gemm_nt_f32_wmma<<<ggrid(NHC, NHC), 256, 0, stream>>>(
        X, NHC, X, NHC, nullptr, 0, Hm, NHC, NHC, NHC, NHC, 1.0f, EPSC);
    // 3. derive Y, D11, C1, RHSB
    derive_kernel<<<dim3(NXC / 256, NXC), 256, 0, stream>>>(Hm, Y1, Chi, Yd, D11, C1, RHSB);
    // 4. Cholesky of P
    cholesky_kernel<<<1, 1024, 0, stream>>>(P, NXC);
    // 5. A = P^-1 Y
    trsm_lower<<<dim3(NXC / 256), 256, 0, stream>>>(P, Yd, Zt, NXC, NXC);
    trsm_upper<<<dim3(NXC / 256), 256, 0, stream>>>(P, Zt, Amat, NXC, NXC);
    // 6. B1 = P^-1 (-H2 - Chi)
    trsm_lower<<<dim3(NQC / 256), 256, 0, stream>>>(P, RHSB, Zt, NXC, NQC);
    trsm_upper<<<dim3(NQC / 256), 256, 0, stream>>>(P, Zt, B1, NXC, NQC);
    // 7. Vx = xi C1^T
    gemm_nt_f32_wmma<<<ggrid(BC, NQC), 256, 0, stream>>>(
        xi, NXC, C1, NXC, nullptr, 0, Vx, NQC, BC, NQC, NXC, 1.0f, 0.0f);
    // 8. blocked tanh scan: w = tanh(Vx + w D11^T) column-block by column-block
    for (int t = 0; t < NQC / KB; ++t) {
        if (t > 0) {
            // Gacc = w[:, :t*KB] * D11[t*KB : t*KB+KB, :t*KB]^T   (WMMA prefix)
            gemm_nt_f32_wmma<<<ggrid(BC, KB), 256, 0, stream>>>(
                w, NQC, D11 + (size_t)t * KB * NQC, NQC, nullptr, 0,
                Gacc, KB, BC, KB, t * KB, 1.0f, 0.0f);
        }
        scan_block_kernel<<<dim3(BC / 256), 256, 0, stream>>>(
            Vx, Gacc, (t > 0) ? 1 : 0, D11, w, t);
    }
    // 9. out = xi A^T + w B1^T
    gemm_nt_f32_wmma<<<ggrid(BC, NXC), 256, 0, stream>>>(
        xi, NXC, Amat, NXC, nullptr, 0, out, NXC, BC, NXC, NXC, 1.0f, 0.0f);
    gemm_nt_f32_wmma<<<ggrid(BC, NXC), 256, 0, stream>>>(
        w, NQC, B1, NQC, out, NXC, out, NXC, BC, NXC, NQC, 1.0f, 0.0f);

    (void)in_sizes; (void)n_in; (void)out_size; (void)ws_size;
}